// RawToRgbTransform_44805098832399
// MI455X (gfx1250) — compile-verified
//
#include <hip/hip_runtime.h>
#include <stdint.h>

namespace {
constexpr int H_IMG = 3072;
constexpr int W_IMG = 4096;
constexpr int TS    = 32;          // output tile edge
constexpr int HALO  = 8;           // full dependency radius of the Menon cascade
constexpr int CW    = TS + 2*HALO; // 48: LDS plane edge
constexpr int NT    = 256;         // 8 waves (wave32)
constexpr float GAMMA_E = 1.0f / 2.2f;
}

#define L(a, y, x) a[(y)*CW + (x)]

__device__ __forceinline__ int reflect_idx(int i, int n) {
  i = (i < 0) ? -i : i;                // jnp 'reflect': -k -> k
  return (i >= n) ? (2*n - 2 - i) : i; // n-1+k -> n-1-k  (halo << n, single fold)
}

// ---- CDNA5 async global->LDS copy path (ASYNCcnt) -------------------------
__device__ __forceinline__ void async_ld_f32(unsigned lds_byte_off, const float* gp) {
  // VGLOBAL async load: vdst = per-lane LDS byte address, vaddr = 64-bit VA, GV mode
  asm volatile("global_load_async_to_lds_b32 %0, %1, off"
               :: "v"(lds_byte_off), "v"(gp)
               : "memory");
}

__device__ __forceinline__ void wait_asynccnt0() {
#if __has_builtin(__builtin_amdgcn_s_wait_asynccnt)
  __builtin_amdgcn_s_wait_asynccnt(0);
#else
  asm volatile("s_wait_asynccnt 0" ::: "memory");
#endif
}

__device__ __forceinline__ void wait_tensorcnt0() {
#if __has_builtin(__builtin_amdgcn_s_wait_tensorcnt)
  __builtin_amdgcn_s_wait_tensorcnt((short)0);
#else
  asm volatile("s_wait_tensorcnt 0" ::: "memory");
#endif
}

// ---- CDNA5 Tensor Data Mover: one 48x48 fp32 tile, global -> packed LDS ---
// D# per CDNA5 ISA. 2-D tensor: groups 2/3 zero-filled.
#if __has_builtin(__builtin_amdgcn_tensor_load_to_lds)
#define HAVE_TDM 1
__device__ __forceinline__ void tdm_load_tile(const float* gbase, unsigned lds_byte_off) {
  typedef unsigned u32x4 __attribute__((ext_vector_type(4)));
  typedef int      i32x8 __attribute__((ext_vector_type(8)));
  typedef int      i32x4 __attribute__((ext_vector_type(4)));

  const unsigned long long ga = (unsigned long long)(uintptr_t)gbase;

  u32x4 g0;
  g0.x = 1u;                                             // count=1, user D#, no gather
  g0.y = lds_byte_off;                                   // lds_addr [63:32]
  g0.z = (unsigned)(ga & 0xFFFFFFFFu);                   // global_addr [95:64]
  g0.w = (unsigned)((ga >> 32) & 0x01FFFFFFu)            // global_addr [120:96]
       | (2u << 30);                                     // type=2 ("image") [127:126]

  i32x8 g1;
  g1[0] = (int)(2u << 16);                               // wg_mask=0, data_size=2 (4B)
  g1[1] = (int)((unsigned)(W_IMG & 0xFFFF) << 16);       // tensor_dim0[15:0] @ [63:48]
  g1[2] = (int)((unsigned)(H_IMG & 0xFFFF) << 16);       // dim0 hi=0 | tensor_dim1 lo
  g1[3] = (int)((unsigned)CW << 16);                     // dim1 hi=0 | tile_dim0=48
  g1[4] = CW;                                            // tile_dim1=48 | tile_dim2=0
  g1[5] = W_IMG;                                         // tensor_dim0_stride[31:0]
  g1[6] = 0;                                             // stride0 hi | stride1 lo
  g1[7] = 0;                                             // stride1 hi

  i32x4 z4 = {0, 0, 0, 0};
#if __clang_major__ >= 23
  i32x8 z8 = {0, 0, 0, 0, 0, 0, 0, 0};
  __builtin_amdgcn_tensor_load_to_lds(g0, g1, z4, z4, z8, 0);
#else
  __builtin_amdgcn_tensor_load_to_lds(g0, g1, z4, z4, 0);
#endif
}
#else
#define HAVE_TDM 0
#endif

// ---------------------------------------------------------------------------
__global__ void init_mmx_kernel(unsigned* __restrict__ mmx) {
  // ordered-uint encoding of non-negative floats: uint compare == float compare
  mmx[0] = 0x7F800000u; // running min, seeded +inf
  mmx[1] = 0x00000000u; // running max, seeded 0.0 (all data >= 0 after clip)
}

// ---------------------------------------------------------------------------
__global__ void __launch_bounds__(NT)
demosaic_kernel(const float* __restrict__ raw, const float* __restrict__ wb,
                float* __restrict__ out, unsigned* __restrict__ mmx) {
  __shared__ float sCFA[CW*CW];
  __shared__ float s1[CW*CW];  // C_H   -> M
  __shared__ float s2[CW*CW];  // C_V   -> G
  __shared__ float s3[CW*CW];  // D_H   -> R-plane
  __shared__ float s4[CW*CW];  // D_V   -> B-plane

  const int tid = threadIdx.x;
  const int gx0 = (int)blockIdx.x * TS - HALO;
  const int gy0 = (int)blockIdx.y * TS - HALO;

  // ---- Stage 1: CFA tile (+halo) HBM -> LDS -------------------------------
  // Interior blocks: one TDM descriptor (TENSORcnt). Image-border blocks:
  // per-element mirror-indexed async loads (ASYNCcnt).
  const bool interior = (gx0 >= 0) && (gy0 >= 0) &&
                        (gx0 + CW <= W_IMG) && (gy0 + CW <= H_IMG);
  bool used_tdm = false;
#if HAVE_TDM
  if (interior) {
    if (tid < 32) {  // one wave issues the DMA (descriptor values are uniform)
      tdm_load_tile(raw + ((size_t)gy0 * W_IMG + gx0),
                    (unsigned)(uintptr_t)&sCFA[0]);
    }
    used_tdm = true;
  }
#endif
  if (!used_tdm) {
    for (int idx = tid; idx < CW*CW; idx += NT) { // 2304 = 9 * 256, no divergence
      int y = idx / CW, x = idx - y*CW;
      int gy = reflect_idx(gy0 + y, H_IMG);
      int gx = reflect_idx(gx0 + x, W_IMG);
      async_ld_f32((unsigned)(uintptr_t)&sCFA[idx], raw + ((size_t)gy * W_IMG + gx));
    }
    wait_asynccnt0();
  }
  wait_tensorcnt0();   // no-op (cnt==0) for waves that issued nothing
  __syncthreads();

  // ---- Stage 2: chroma residues C_H (s1), C_V (s2) on halo-6 region ------
  float* sCH = s1; float* sCV = s2;
  for (int idx = tid; idx < 44*44; idx += NT) {
    int y = idx / 44 + 2, x = idx % 44 + 2;
    int gy = gy0 + y, gx = gx0 + x;              // parity survives mirroring (H,W even)
    float c = L(sCFA, y, x);
    if (((gy ^ gx) & 1) != 0) {                  // green site: C = 0
      L(sCH, y, x) = 0.0f;
      L(sCV, y, x) = 0.0f;
    } else {                                     // h0+h1 directional green estimates
      float gh = 0.5f*(L(sCFA,y,x-1) + L(sCFA,y,x+1)) + 0.5f*c
               - 0.25f*(L(sCFA,y,x-2) + L(sCFA,y,x+2));
      float gv = 0.5f*(L(sCFA,y-1,x) + L(sCFA,y+1,x)) + 0.5f*c
               - 0.25f*(L(sCFA,y-2,x) + L(sCFA,y+2,x));
      L(sCH, y, x) = c - gh;
      L(sCV, y, x) = c - gv;
    }
  }
  __syncthreads();

  // ---- Stage 3: gradients D_H (s3), D_V (s4) on halo-4 region ------------
  float* sDH = s3; float* sDV = s4;
  for (int idx = tid; idx < 40*40; idx += NT) {
    int y = idx / 40 + 4, x = idx % 40 + 4;
    L(sDH, y, x) = fabsf(L(sCH, y, x) - L(sCH, y, x+2));
    L(sDV, y, x) = fabsf(L(sCV, y, x) - L(sCV, y+2, x));
  }
  __syncthreads();

  // ---- Stage 4: classifier M (s1) + green plane G (s2) on halo-2 ---------
  // d = conv(k_row,x) + conv(k_col,y) - 3*D  =  3D(y,x)+3D(y,x+2)+D(y-2,x)+D(y+2,x)
  // (safe to overwrite C planes: this stage reads only D and CFA)
  float* sM = s1; float* sG = s2;
  for (int idx = tid; idx < 36*36; idx += NT) {
    int y = idx / 36 + 6, x = idx % 36 + 6;
    int gy = gy0 + y, gx = gx0 + x;
    float dH = 3.0f*L(sDH,y,x) + 3.0f*L(sDH,y,x+2) + L(sDH,y-2,x) + L(sDH,y+2,x);
    float dV = 3.0f*L(sDV,y,x) + 3.0f*L(sDV,y,x+2) + L(sDV,y-2,x) + L(sDV,y+2,x);
    float c = L(sCFA, y, x);
    float g;
    if (((gy ^ gx) & 1) != 0) {
      g = c;
    } else {
      float gh = 0.5f*(L(sCFA,y,x-1) + L(sCFA,y,x+1)) + 0.5f*c
               - 0.25f*(L(sCFA,y,x-2) + L(sCFA,y,x+2));
      float gv = 0.5f*(L(sCFA,y-1,x) + L(sCFA,y+1,x)) + 0.5f*c
               - 0.25f*(L(sCFA,y-2,x) + L(sCFA,y+2,x));
      g = (dV >= dH) ? gh : gv;
    }
    L(sM, y, x) = (dV >= dH) ? 1.0f : 0.0f;
    L(sG, y, x) = g;
  }
  __syncthreads();

  // ---- Stage 5: R (s3) / B (s4) planes at green sites on halo-1 ----------
  // (safe to overwrite D planes: this stage reads only CFA and G)
  float* sRp = s3; float* sBp = s4;
  for (int idx = tid; idx < 34*34; idx += NT) {
    int y = idx / 34 + 7, x = idx % 34 + 7;
    int gy = gy0 + y, gx = gx0 + x;
    int py = gy & 1, px = gx & 1;                // RGGB: (0,0)=R (0,1)/(1,0)=G (1,1)=B
    float c = L(sCFA, y, x);
    float r, b;
    if ((py | px) == 0)      { r = c;    b = 0.0f; }  // red site (B filled stage 6)
    else if ((py & px) == 1) { b = c;    r = 0.0f; }  // blue site
    else {
      float g  = L(sG, y, x);
      float hh = g + 0.5f*(L(sCFA,y,x-1)+L(sCFA,y,x+1)) - 0.5f*(L(sG,y,x-1)+L(sG,y,x+1));
      float vv = g + 0.5f*(L(sCFA,y-1,x)+L(sCFA,y+1,x)) - 0.5f*(L(sG,y-1,x)+L(sG,y+1,x));
      if (py == 0) { r = hh; b = vv; }           // green in red row
      else         { b = hh; r = vv; }           // green in blue row
    }
    L(sRp, y, x) = r;
    L(sBp, y, x) = b;
  }
  __syncthreads();

  // ---- Stage 6: chroma at opposite sites, WB (with 1/65535 folded), clip,
  //               store planes, accumulate block min/max --------------------
  const float wscale = 1.0f / 65535.0f;
  const float wbR = wb[0] * wscale;
  const float wbG = wb[1] * wscale;
  const float wbB = wb[2] * wscale;

  float mn = 3.0e38f;
  float mx = 0.0f;

  for (int idx = tid; idx < TS*TS; idx += NT) {  // 4 pixels / thread
    int y = idx / TS + HALO, x = idx % TS + HALO;
    int gy = gy0 + y, gx = gx0 + x;
    if (gy < H_IMG && gx < W_IMG) {
      int py = gy & 1, px = gx & 1;
      float c = L(sCFA, y, x);
      float R, G, B;
      if (((py ^ px) & 1) != 0) {                // green site
        G = c; R = L(sRp, y, x); B = L(sBp, y, x);
      } else {
        float m = L(sM, y, x);
        G = L(sG, y, x);
        float hR = 0.5f*(L(sRp,y,x-1) + L(sRp,y,x+1));
        float hB = 0.5f*(L(sBp,y,x-1) + L(sBp,y,x+1));
        float vR = 0.5f*(L(sRp,y-1,x) + L(sRp,y+1,x));
        float vB = 0.5f*(L(sBp,y-1,x) + L(sBp,y+1,x));
        if (py == 0) {                           // red site: interpolate B
          R = c;
          B = m*(c + hB - hR) + (1.0f - m)*(c + vB - vR);
        } else {                                 // blue site: interpolate R
          B = c;
          R = m*(c + hR - hB) + (1.0f - m)*(c + vR - vB);
        }
      }
      R = fminf(fmaxf(R * wbR, 0.0f), 1.0f);
      G = fminf(fmaxf(G * wbG, 0.0f), 1.0f);
      B = fminf(fmaxf(B * wbB, 0.0f), 1.0f);
      int o = gy * W_IMG + gx;
      out[o]                   = R;   // RT hint: RGB (151 MB) fits in 192 MB L2,
      out[H_IMG*W_IMG + o]     = G;   // finalize pass re-reads from L2
      out[2*H_IMG*W_IMG + o]   = B;
      mn = fminf(mn, fminf(R, fminf(G, B)));
      mx = fmaxf(mx, fmaxf(R, fmaxf(G, B)));
    }
  }

  // wave32 butterfly reduction, then one ordered-uint atomic per wave
  for (int m = 16; m > 0; m >>= 1) {
    mn = fminf(mn, __shfl_xor(mn, m, 32));
    mx = fmaxf(mx, __shfl_xor(mx, m, 32));
  }
  if ((tid & 31) == 0) {
    atomicMin(&mmx[0], __float_as_uint(mn));
    atomicMax(&mmx[1], __float_as_uint(mx));
  }
}

// ---------------------------------------------------------------------------
__global__ void __launch_bounds__(NT)
finalize_kernel(float* __restrict__ out, const unsigned* __restrict__ mmx) {
  const int n4 = (3 * H_IMG * W_IMG) / 4;        // exactly divisible
  int i = (int)blockIdx.x * NT + threadIdx.x;
  if (i >= n4) return;
  const float mn  = __uint_as_float(mmx[0]);
  const float mx  = __uint_as_float(mmx[1]);
  const float rng = mx - mn;
  const bool  donorm = rng > 1e-6f;
  const float inv = 1.0f / fmaxf(rng, 1e-12f);
  float4 v = reinterpret_cast<float4*>(out)[i];
  float a[4] = {v.x, v.y, v.z, v.w};
#pragma unroll
  for (int k = 0; k < 4; ++k) {
    float x = a[k];
    if (donorm) x = (x - mn) * inv;
    a[k] = powf(fmaxf(x, 1e-12f), GAMMA_E);
  }
  v.x = a[0]; v.y = a[1]; v.z = a[2]; v.w = a[3];
  reinterpret_cast<float4*>(out)[i] = v;
}

// ---------------------------------------------------------------------------
extern "C" void kernel_launch(void* const* d_in, const int* in_sizes, int n_in,
                              void* d_out, int out_size, void* d_ws, size_t ws_size,
                              hipStream_t stream) {
  const float* raw = (const float*)d_in[0];   // (H, W) fp32
  const float* wb  = (const float*)d_in[1];   // (3,) fp32
  float*    out = (float*)d_out;              // (3, H, W) fp32
  unsigned* mmx = (unsigned*)d_ws;            // [min_bits, max_bits]

  init_mmx_kernel<<<1, 1, 0, stream>>>(mmx);

  dim3 grid(W_IMG / TS, H_IMG / TS);          // 128 x 96 tiles, exact
  demosaic_kernel<<<grid, NT, 0, stream>>>(raw, wb, out, mmx);

  const int n4 = (3 * H_IMG * W_IMG) / 4;
  finalize_kernel<<<(n4 + NT - 1) / NT, NT, 0, stream>>>(out, mmx);

  (void)in_sizes; (void)n_in; (void)out_size; (void)ws_size;
}